// DyReLUB_35622458753569
// MI455X (gfx1250) — compile-verified
//
#include <hip/hip_runtime.h>
#include <math.h>

typedef float v2f __attribute__((ext_vector_type(2)));
typedef float v8f __attribute__((ext_vector_type(8)));

constexpr int C    = 256;   // channels
constexpr int HID  = 64;    // C / reduction
constexpr int OUT2 = 1024;  // 2*K*C

// ---------------- helper kernels ----------------

__global__ void zero_kernel(float* p, int n) {
  int i = blockIdx.x * blockDim.x + threadIdx.x;
  if (i < n) p[i] = 0.0f;
}

// deg[n] += 1 for each edge with dst==n (self-loop +1 added in dis_kernel)
__global__ void deg_kernel(const int* __restrict__ dst, int E, float* __restrict__ deg) {
  int i = blockIdx.x * blockDim.x + threadIdx.x;
  if (i < E) atomicAdd(&deg[dst[i]], 1.0f);
}

// in place: deg -> dis = rsqrt(deg + 1)
__global__ void dis_kernel(float* __restrict__ deg, int N) {
  int i = blockIdx.x * blockDim.x + threadIdx.x;
  if (i < N) deg[i] = rsqrtf(deg[i] + 1.0f);
}

// w[src_e] += dis[dst_e]
__global__ void w_kernel(const int* __restrict__ src, const int* __restrict__ dst, int E,
                         const float* __restrict__ dis, float* __restrict__ w) {
  int i = blockIdx.x * blockDim.x + threadIdx.x;
  if (i < E) atomicAdd(&w[src[i]], dis[dst[i]]);
}

// s[c] = sum_n wn[n] * x[n,c],  wn[n] = dis[n]*(w[n]+dis[n])
// block = 256 threads, 128 rows per block, float4 over channels
__global__ void s_kernel(const float* __restrict__ x, const float* __restrict__ dis,
                         const float* __restrict__ w, int N, float* __restrict__ s) {
  __shared__ float wn[128];
  __shared__ float part[1024]; // [g(4)][c4(64)][j(4)]
  int t  = threadIdx.x;
  int r0 = blockIdx.x * 128;
  if (t < 128) {
    int r = r0 + t;
    float v = 0.0f;
    if (r < N) { float d = dis[r]; v = d * (w[r] + d); }
    wn[t] = v;
  }
  __syncthreads();
  int g  = t >> 6;   // 0..3 (row sub-group)
  int c4 = t & 63;   // float4 channel group
  const float4* x4 = (const float4*)x;
  float4 acc = make_float4(0.f, 0.f, 0.f, 0.f);
  for (int rr = g; rr < 128; rr += 4) {
    int r = r0 + rr;
    if (r >= N) break;
    float  wv = wn[rr];
    float4 xv = x4[(size_t)r * 64 + c4];
    acc.x = fmaf(wv, xv.x, acc.x);
    acc.y = fmaf(wv, xv.y, acc.y);
    acc.z = fmaf(wv, xv.z, acc.z);
    acc.w = fmaf(wv, xv.w, acc.w);
  }
  int pi = (g * 64 + c4) * 4;
  part[pi + 0] = acc.x; part[pi + 1] = acc.y;
  part[pi + 2] = acc.z; part[pi + 3] = acc.w;
  __syncthreads();
  if (t < 64) {
    for (int j = 0; j < 4; ++j) {
      float v = part[(0 * 64 + t) * 4 + j] + part[(1 * 64 + t) * 4 + j] +
                part[(2 * 64 + t) * 4 + j] + part[(3 * 64 + t) * 4 + j];
      atomicAdd(&s[t * 4 + j], v);
    }
  }
}

// ---------------- WMMA coefficient kernel (single block, 8 waves) ----------------
// t = relu(s/N @ W1 + b1)  : 1x256 @ 256x64  via V_WMMA_F32_16X16X4_F32
// th2 = t @ W2 + b2        : 1x64  @ 64x1024 via V_WMMA_F32_16X16X4_F32
// coefs[col] = tanh(th2/2)*lambda[col%4] + init[col%4]
__global__ void coef_kernel(const float* __restrict__ s, const float* __restrict__ W1,
                            const float* __restrict__ b1, const float* __restrict__ W2,
                            const float* __restrict__ b2, float* __restrict__ coefs,
                            float invN) {
  __shared__ float sN[C];
  __shared__ float tl[HID];
  int t = threadIdx.x;         // 256 threads = 8 waves (wave32)
  sN[t] = s[t] * invN;
  __syncthreads();

  int wave = t >> 5;
  int lane = t & 31;
  int half = lane >> 4;        // 0: K+{0,1}, 1: K+{2,3}
  int idx  = lane & 15;        // N within tile (and M for A, unused)

  // GEMM1: waves 0..3 each produce one 16-col tile of t
  if (wave < 4) {
    int col = wave * 16 + idx;
    v8f c = {};
    for (int k0 = 0; k0 < C; k0 += 4) {
      v2f a, b;
      a.x = sN[k0 + 2 * half];             // all 16 A-rows broadcast = sN
      a.y = sN[k0 + 2 * half + 1];
      b.x = W1[(k0 + 2 * half) * HID + col];
      b.y = W1[(k0 + 2 * half + 1) * HID + col];
      c = __builtin_amdgcn_wmma_f32_16x16x4_f32(false, a, false, b,
                                                (short)0, c, false, false);
    }
    if (lane < 16) {                       // row M=0 of D lives in c[0], lanes 0..15
      float v = c[0] + b1[col];
      tl[col] = v > 0.0f ? v : 0.0f;
    }
  }
  __syncthreads();

  // GEMM2: 64 N-tiles over 8 waves
  for (int tile = wave; tile < 64; tile += 8) {
    int col = tile * 16 + idx;
    v8f c = {};
    for (int k0 = 0; k0 < HID; k0 += 4) {
      v2f a, b;
      a.x = tl[k0 + 2 * half];
      a.y = tl[k0 + 2 * half + 1];
      b.x = W2[(k0 + 2 * half) * OUT2 + col];
      b.y = W2[(k0 + 2 * half + 1) * OUT2 + col];
      c = __builtin_amdgcn_wmma_f32_16x16x4_f32(false, a, false, b,
                                                (short)0, c, false, false);
    }
    if (lane < 16) {
      float th  = c[0] + b2[col];
      float g   = tanhf(0.5f * th);        // == 2*sigmoid(th) - 1
      int   j   = col & 3;
      float lam = (j < 2) ? 1.0f : 0.5f;
      float ini = (j == 0) ? 1.0f : 0.0f;
      coefs[col] = fmaf(g, lam, ini);
    }
  }
}

// ---------------- final elementwise max kernel ----------------
__global__ void out_kernel(const float4* __restrict__ x4, const float* __restrict__ coefs,
                           float4* __restrict__ out4, int n4) {
  __shared__ float cf[1024];
  int t = threadIdx.x;
  for (int j = t; j < 1024; j += 256) cf[j] = coefs[j];
  __syncthreads();
  int i = blockIdx.x * 256 + t;
  if (i >= n4) return;
  float4 xv = x4[i];
  int cg = (i & 63) << 2;  // base channel of this float4 (256 ch = 64 groups/row)
  float4 o;
  {
    int b = (cg + 0) << 2;
    o.x = fmaxf(fmaf(xv.x, cf[b + 0], cf[b + 2]), fmaf(xv.x, cf[b + 1], cf[b + 3]));
  }
  {
    int b = (cg + 1) << 2;
    o.y = fmaxf(fmaf(xv.y, cf[b + 0], cf[b + 2]), fmaf(xv.y, cf[b + 1], cf[b + 3]));
  }
  {
    int b = (cg + 2) << 2;
    o.z = fmaxf(fmaf(xv.z, cf[b + 0], cf[b + 2]), fmaf(xv.z, cf[b + 1], cf[b + 3]));
  }
  {
    int b = (cg + 3) << 2;
    o.w = fmaxf(fmaf(xv.w, cf[b + 0], cf[b + 2]), fmaf(xv.w, cf[b + 1], cf[b + 3]));
  }
  out4[i] = o;
}

// ---------------- launch ----------------
extern "C" void kernel_launch(void* const* d_in, const int* in_sizes, int n_in,
                              void* d_out, int out_size, void* d_ws, size_t ws_size,
                              hipStream_t stream) {
  const float* x  = (const float*)d_in[0];
  const int*   ei = (const int*)d_in[1];   // edge_index [2, E] (int32 under default JAX x64-off)
  const float* W1 = (const float*)d_in[2];
  const float* b1 = (const float*)d_in[3];
  const float* W2 = (const float*)d_in[4];
  const float* b2 = (const float*)d_in[5];
  float* out = (float*)d_out;

  int N = in_sizes[0] / C;   // 100000
  int E = in_sizes[1] / 2;   // 3200000
  const int* src = ei;
  const int* dst = ei + E;

  float* ws    = (float*)d_ws;
  float* deg   = ws;                 // N floats (becomes dis in place)
  float* w     = ws + N;             // N floats
  float* s     = ws + 2 * N;         // 256 floats
  float* coefs = ws + 2 * N + C;     // 1024 floats

  int nz = 2 * N + C;
  zero_kernel<<<(nz + 255) / 256, 256, 0, stream>>>(ws, nz);
  deg_kernel<<<(E + 255) / 256, 256, 0, stream>>>(dst, E, deg);
  dis_kernel<<<(N + 255) / 256, 256, 0, stream>>>(deg, N);
  w_kernel  <<<(E + 255) / 256, 256, 0, stream>>>(src, dst, E, deg, w);
  s_kernel  <<<(N + 127) / 128, 256, 0, stream>>>(x, deg, w, N, s);
  coef_kernel<<<1, 256, 0, stream>>>(s, W1, b1, W2, b2, coefs, 1.0f / (float)N);

  int n4 = (N * C) / 4;
  out_kernel<<<(n4 + 255) / 256, 256, 0, stream>>>((const float4*)x, coefs, (float4*)out, n4);
}